// TokenGuideLoss_44959717655166
// MI455X (gfx1250) — compile-verified
//
#include <hip/hip_runtime.h>
#include <hip/hip_bf16.h>
#include <math.h>

// ---------------------------------------------------------------------------
// Problem constants
// ---------------------------------------------------------------------------
#define B_TOTAL   524288
#define EPT       4                         // elements per thread (16B-aligned 1296B chunk of dir_9_d)
#define TPB       256
#define GRID1     (B_TOTAL / (EPT * TPB))   // 512 blocks
#define LOSS_EPS  1e-7f
#define COS_EPS_  1e-8f

typedef __attribute__((ext_vector_type(2))) float v2f;
typedef __attribute__((ext_vector_type(8))) float v8f;

__device__ __forceinline__ float clip_acos(float x) {
    const float lo = -1.0f + LOSS_EPS;
    const float hi =  1.0f - LOSS_EPS;
    return acosf(fminf(fmaxf(x, lo), hi));
}

// ---------------------------------------------------------------------------
// Pass 1: per-thread 4 elements, streamed b128 loads, per-block partial sums.
// part layout (SoA): part[q * GRID1 + blockIdx.x], q = {pred, ori, pre, our}
// ---------------------------------------------------------------------------
__global__ __launch_bounds__(TPB)
void tgl_partial_kernel(const float* __restrict__ m1,
                        const float* __restrict__ m2,
                        const float* __restrict__ lab,
                        const float* __restrict__ dir,
                        float* __restrict__ part)
{
    const int tid = blockIdx.x * TPB + threadIdx.x;      // 0 .. 131071

    const float4* m1v  = (const float4*)m1  + (size_t)tid * 9;   // 36 floats, 16B aligned
    const float4* m2v  = (const float4*)m2  + (size_t)tid * 9;
    const float4* labv = (const float4*)lab + (size_t)tid * 3;   // 12 floats
    const float4* dirv = (const float4*)dir + (size_t)tid * 81;  // 324 floats

    // --- load m1 / m2 for the 4 elements (aligned float4) ---
    float m1f[36], m2f[36];
#pragma unroll
    for (int i = 0; i < 9; ++i) {
        float4 a = m1v[i];
        m1f[4*i+0] = a.x; m1f[4*i+1] = a.y; m1f[4*i+2] = a.z; m1f[4*i+3] = a.w;
        float4 b = m2v[i];
        m2f[4*i+0] = b.x; m2f[4*i+1] = b.y; m2f[4*i+2] = b.z; m2f[4*i+3] = b.w;
    }
    float labf[12];
#pragma unroll
    for (int i = 0; i < 3; ++i) {
        float4 v = labv[i];
        labf[4*i+0] = v.x; labf[4*i+1] = v.y; labf[4*i+2] = v.z; labf[4*i+3] = v.w;
    }

    // --- stream dir_9_d: accumulate sum_j dot9(m1, dir[e,j]) per element ---
    float d1o[EPT] = {0.f, 0.f, 0.f, 0.f};
#pragma unroll
    for (int f4 = 0; f4 < 81; ++f4) {
        float4 v = dirv[f4];
#pragma unroll
        for (int c = 0; c < 4; ++c) {
            const int flat = f4 * 4 + c;      // e*81 + j*9 + k (constants after unroll)
            const int e    = flat / 81;
            const int k    = flat % 9;
            const float comp = (c == 0) ? v.x : (c == 1) ? v.y : (c == 2) ? v.z : v.w;
            d1o[e] = fmaf(comp, m1f[e * 9 + k], d1o[e]);
        }
    }

    // --- per-element losses ---
    float acc[4] = {0.f, 0.f, 0.f, 0.f};     // pred, ori, pre, our
#pragma unroll
    for (int e = 0; e < EPT; ++e) {
        float d12 = 0.f, n1s = 0.f, n2s = 0.f;
#pragma unroll
        for (int k = 0; k < 9; ++k) {
            const float a = m1f[e * 9 + k];
            const float b = m2f[e * 9 + k];
            d12 = fmaf(a, b, d12);
            n1s = fmaf(a, a, n1s);
            n2s = fmaf(b, b, n2s);
        }
        const float pitch = labf[e * 3 + 0];
        const float yaw   = labf[e * 3 + 1];
        const int yaw_bin   = (yaw   >  40.0f) ? 0 : ((yaw   >= -40.0f) ? 1 : 2);
        const int pitch_bin = (pitch >  30.0f) ? 0 : ((pitch >= -30.0f) ? 1 : 2);
        const int loc = pitch_bin * 3 + yaw_bin;

        // region-selected prediction: 9 scalar loads (hot in WGP$ from the stream above)
        const float* rg = dir + (size_t)(tid * EPT + e) * 81 + loc * 9;
        float d1r = 0.f;
#pragma unroll
        for (int k = 0; k < 9; ++k) d1r = fmaf(m1f[e * 9 + k], rg[k], d1r);

        const float n1 = fmaxf(sqrtf(n1s), COS_EPS_);
        const float n2 = fmaxf(sqrtf(n2s), COS_EPS_);

        acc[0] += clip_acos((d12 - 1.0f) * 0.5f);
        acc[1] += clip_acos((d1r - 1.0f) * 0.5f);
        acc[2] += clip_acos(d12 / (n1 * n2));
        acc[3] += clip_acos((d1o[e] * (1.0f / 9.0f) - 1.0f) * 0.5f);
    }

    // --- deterministic block reduction: wave32 shuffle tree + LDS stage ---
#pragma unroll
    for (int q = 0; q < 4; ++q) {
        float v = acc[q];
#pragma unroll
        for (int off = 16; off >= 1; off >>= 1) v += __shfl_xor(v, off, 32);
        acc[q] = v;
    }
    __shared__ float s[TPB / 32][4];
    const int wave = threadIdx.x >> 5;
    if ((threadIdx.x & 31) == 0) {
#pragma unroll
        for (int q = 0; q < 4; ++q) s[wave][q] = acc[q];
    }
    __syncthreads();
    if (threadIdx.x < 4) {
        float t = 0.f;
#pragma unroll
        for (int w = 0; w < TPB / 32; ++w) t += s[w][threadIdx.x];
        part[threadIdx.x * GRID1 + blockIdx.x] = t;   // fully overwritten every call
    }
}

// ---------------------------------------------------------------------------
// Pass 2: single wave. Reduce 512 partials per accumulator with
// v_wmma_f32_16x16x4_f32 (A = ones -> exact f32 add-reduction of all 64
// B-operand values per issue, accumulator-chained through C), then a 16-lane
// shuffle fold. Deterministic, f32-exact.
// ---------------------------------------------------------------------------
__global__ __launch_bounds__(32)
void tgl_final_kernel(const float* __restrict__ part, float* __restrict__ out)
{
    const int lane = threadIdx.x;

    v2f ones;
    ones.x = 1.0f;
    ones.y = 1.0f;

    float tot[4];
#pragma unroll
    for (int q = 0; q < 4; ++q) {
        v8f c = {};
#pragma unroll
        for (int it = 0; it < GRID1 / 64; ++it) {        // 8 WMMAs of 64 values
            v2f b;
            b.x = part[q * GRID1 + it * 64 + lane];
            b.y = part[q * GRID1 + it * 64 + 32 + lane];
            // D[m,n] = sum_k 1 * B[k,n] + C[m,n] -> every loaded value lands in
            // exactly one matrix cell; row sums give the grand total.
            c = __builtin_amdgcn_wmma_f32_16x16x4_f32(
                    /*neg_a=*/false, ones, /*neg_b=*/false, b,
                    /*c_mod=*/(short)0, c, /*reuse_a=*/false, /*reuse_b=*/false);
        }
        // Row M=0 lives in VGPR0 of D across lanes 0..15 (lanes 16..31 hold the
        // identical row M=8). Fold the 16-lane group.
        float v = c[0];
        v += __shfl_xor(v, 8, 32);
        v += __shfl_xor(v, 4, 32);
        v += __shfl_xor(v, 2, 32);
        v += __shfl_xor(v, 1, 32);
        tot[q] = v;
    }

    if (lane == 0) {
        const float inv = 1.0f / (float)B_TOTAL;
        const float pred = tot[0] * inv;
        const float ori  = tot[1] * inv;
        const float pre  = tot[2] * inv;
        const float our  = tot[3] * inv;
        out[0] = 0.6f * pred + 0.4f * ori + 0.1f * pre + 0.1f * our;
        out[1] = pred;
        out[2] = ori;
        out[3] = pre;
        out[4] = our;
    }
}

// ---------------------------------------------------------------------------
// Launcher
// ---------------------------------------------------------------------------
extern "C" void kernel_launch(void* const* d_in, const int* in_sizes, int n_in,
                              void* d_out, int out_size, void* d_ws, size_t ws_size,
                              hipStream_t stream)
{
    const float* m1  = (const float*)d_in[0];   // (B,3,3)
    const float* m2  = (const float*)d_in[1];   // (B,3,3)
    const float* lab = (const float*)d_in[2];   // (B,3)
    const float* dir = (const float*)d_in[3];   // (B,9,3,3)
    float* out  = (float*)d_out;                // 5 scalars
    float* part = (float*)d_ws;                 // 4 * GRID1 floats (8 KB)

    tgl_partial_kernel<<<GRID1, TPB, 0, stream>>>(m1, m2, lab, dir, part);
    tgl_final_kernel<<<1, 32, 0, stream>>>(part, out);
}